// SamplingAggregator_17824114279119
// MI455X (gfx1250) — compile-verified
//
#include <hip/hip_runtime.h>

typedef __attribute__((ext_vector_type(2))) float v2f;
typedef __attribute__((ext_vector_type(8))) float v8f;

#define D_     128
#define K_     32
#define HID_   100
#define OUT_   20
#define H_     5
#define OUTROW (H_*OUT_ + D_)   // 228
#define HSTR   260              // h_lds row stride (floats): 260 mod 64 = 4 -> conflict-free A reads
#define S1     108              // out1_lds row stride: 108 mod 64 = 44, 16 rows hit distinct banks
#define TS     24               // t_lds row stride

__device__ __forceinline__ v8f wmma_f32_k4(v2f a, v2f b, v8f c) {
  // D = A(16x4 f32) * B(4x16 f32) + C(16x16 f32)
  return __builtin_amdgcn_wmma_f32_16x16x4_f32(
      /*neg_a=*/false, a, /*neg_b=*/false, b,
      /*c_mod=*/(short)0, c, /*reuse_a=*/false, /*reuse_b=*/false);
}

__global__ __launch_bounds__(256)
void sag_fused_kernel(const float* __restrict__ emb,
                      const int*   __restrict__ node_idx,
                      const int*   __restrict__ neigh_idx,
                      const float* __restrict__ W1, const float* __restrict__ b1,
                      const float* __restrict__ W2, const float* __restrict__ b2,
                      const float* __restrict__ Wa, const float* __restrict__ ba,
                      float* __restrict__ out)
{
  __shared__ float h_lds[K_ * HSTR];   // 32 x 256 (padded): [neigh_emb | node_emb]
  __shared__ float o1_lds[K_ * S1];    // 32 x 100 (padded): relu(h @ W1 + b1)
  __shared__ float t_lds[K_ * TS];     // 32 x 20  (padded): relu(o1 @ W2 + b2)
  __shared__ float att_lds[K_ * H_];   // 32 x 5
  __shared__ float p_lds[K_ * H_];     // softmax(att) flat [k*5+h]
  __shared__ int   idx_s[K_ + 1];

  const int tid  = threadIdx.x;
  const int lane = tid & 31;
  const int w    = tid >> 5;
  const int blk  = blockIdx.x;

  if (tid < K_)  idx_s[tid] = neigh_idx[(size_t)blk * K_ + tid];
  if (tid == K_) idx_s[K_]  = node_idx[blk];
  __syncthreads();

  // ---- Phase 0: gather h rows into LDS (float4-coalesced; L2-resident table) ----
  #pragma unroll
  for (int it = 0; it < 8; ++it) {
    int i   = tid + 256 * it;          // 2048 float4 slots = 32 rows x 64
    int row = i >> 6;
    int q   = i & 63;                  // float4 index in 256-float row
    int src = (q < 32) ? idx_s[row] : idx_s[K_];
    int qq  = (q < 32) ? q : (q - 32);
    float4 v = ((const float4*)(emb + (size_t)src * D_))[qq];
    *(float4*)&h_lds[row * HSTR + q * 4] = v;
  }
  __syncthreads();

  // ---- Phase 1: GEMM1  (32 x 256) @ (256 x 100), peek-zero rows of W1 pruned ----
  if (w < 7) {
    const int  c  = w * 16 + (lane & 15);   // output column (<=111; W1 reads stay in bounds)
    const bool cv = (c < HID_);
    const int  kr = 2 * (lane >> 4);        // 0 or 2 (A/B fragment K sub-offset)
    const int  m  = lane & 15;
    v8f acc0 = {}; v8f acc1 = {};
    const float* w1c = W1 + c;
    // Unpredicated B loads: lanes with c>=100 feed only dead D columns (never
    // stored), and max index 255*100+111 is within W1's 306*100 extent.
    #pragma unroll 8
    for (int k0 = 0; k0 < 2 * D_; k0 += 4) {
      int r = k0 + kr;
      v2f a0 = *(const v2f*)&h_lds[m * HSTR + r];          // M-tile 0 (pairs 0..15)
      v2f a1 = *(const v2f*)&h_lds[(m + 16) * HSTR + r];   // M-tile 1 (pairs 16..31)
      v2f b;
      b.x = w1c[r * HID_];
      b.y = w1c[(r + 1) * HID_];
      acc0 = wmma_f32_k4(a0, b, acc0);
      acc1 = wmma_f32_k4(a1, b, acc1);
    }
    if (cv) {
      float bb = b1[c];
      #pragma unroll
      for (int v = 0; v < 8; ++v) {
        int M = v + 8 * (lane >> 4);
        float x0 = acc0[v] + bb;
        float x1 = acc1[v] + bb;
        o1_lds[M * S1 + c]        = x0 > 0.0f ? x0 : 0.0f;
        o1_lds[(M + 16) * S1 + c] = x1 > 0.0f ? x1 : 0.0f;
      }
    }
  }
  __syncthreads();

  // ---- Phase 2: GEMM2  (32 x 100) @ (100 x 20) ----
  if (w < 4) {
    const int  mt = w >> 1, nt = w & 1;
    const int  n  = nt * 16 + (lane & 15);
    const bool nv = (n < OUT_);
    const int  nc = nv ? n : (OUT_ - 1);    // clamped column: stays in-bounds, dead lanes
    const int  kr = 2 * (lane >> 4);        // compute duplicates of col 19 (never stored)
    const int  m  = (lane & 15) + 16 * mt;
    v8f acc = {};
    #pragma unroll 5
    for (int k0 = 0; k0 < HID_; k0 += 4) {
      int r = k0 + kr;
      v2f a = *(const v2f*)&o1_lds[m * S1 + r];
      v2f b;
      b.x = W2[r * OUT_ + nc];
      b.y = W2[(r + 1) * OUT_ + nc];
      acc = wmma_f32_k4(a, b, acc);
    }
    if (nv) {
      float bb = b2[n];
      #pragma unroll
      for (int v = 0; v < 8; ++v) {
        int M = v + 8 * (lane >> 4) + 16 * mt;
        float x = acc[v] + bb;
        t_lds[M * TS + n] = x > 0.0f ? x : 0.0f;
      }
    }
  }
  __syncthreads();

  // ---- Phase 3: attention heads + softmax over H ----
  if (tid < K_ * H_) {
    int k = tid / H_, h = tid % H_;
    float s = ba[h];
    #pragma unroll
    for (int o = 0; o < OUT_; ++o) s += t_lds[k * TS + o] * Wa[o * H_ + h];
    att_lds[tid] = s > 0.0f ? s : 0.0f;
  }
  __syncthreads();
  if (tid < K_) {
    float m0 = att_lds[tid * H_];
    #pragma unroll
    for (int h = 1; h < H_; ++h) m0 = fmaxf(m0, att_lds[tid * H_ + h]);
    float e[H_], s = 0.0f;
    #pragma unroll
    for (int h = 0; h < H_; ++h) { e[h] = __expf(att_lds[tid * H_ + h] - m0); s += e[h]; }
    float inv = 1.0f / s;
    #pragma unroll
    for (int h = 0; h < H_; ++h) p_lds[tid * H_ + h] = e[h] * inv;
  }
  __syncthreads();

  // ---- Phase 4: scrambled-reshape weighted aggregation (faithful to reference) ----
  float* orow = out + (size_t)blk * OUTROW;
  if (tid < H_ * OUT_) {
    int hh = tid / OUT_, o = tid % OUT_;
    float s = 0.0f;
    #pragma unroll
    for (int kk = 0; kk < K_; ++kk)
      s += p_lds[hh * K_ + kk] * t_lds[kk * TS + o];   // a[h,k] = p_flat[h*32+k]
    orow[tid] = s;
  }
  // ---- Phase 5: append node embedding (cols 128..255 of any h row) ----
  if (tid < D_) orow[H_ * OUT_ + tid] = h_lds[D_ + tid];
}

extern "C" void kernel_launch(void* const* d_in, const int* in_sizes, int n_in,
                              void* d_out, int out_size, void* d_ws, size_t ws_size,
                              hipStream_t stream) {
  const float* emb       = (const float*)d_in[0];
  const int*   node_idx  = (const int*)  d_in[1];
  const int*   neigh_idx = (const int*)  d_in[2];
  const float* W1        = (const float*)d_in[3];
  const float* b1        = (const float*)d_in[4];
  const float* W2        = (const float*)d_in[5];
  const float* b2        = (const float*)d_in[6];
  const float* Wa        = (const float*)d_in[7];
  const float* ba        = (const float*)d_in[8];
  float* out = (float*)d_out;
  const int N = in_sizes[1];   // number of nodes (node_idx length)

  sag_fused_kernel<<<dim3(N), dim3(256), 0, stream>>>(
      emb, node_idx, neigh_idx, W1, b1, W2, b2, Wa, ba, out);
}